// REALS_model_multi_62105227100312
// MI455X (gfx1250) — compile-verified
//
#include <hip/hip_runtime.h>
#include <hip/hip_bf16.h>
#include <math.h>

typedef __attribute__((ext_vector_type(2))) float v2f;
typedef __attribute__((ext_vector_type(8))) float v8f;

#define T_DIM 256

// ---------------------------------------------------------------------------
// zero the per-level dot-product accumulators (4 levels x 256 t)
// ---------------------------------------------------------------------------
__global__ void zero_kernel(float* __restrict__ p, int n) {
    int i = blockIdx.x * blockDim.x + threadIdx.x;
    if (i < n) p[i] = 0.0f;
}

// ---------------------------------------------------------------------------
// 2x2 average pool: in (T, 2Ho, 2Wo) -> out (T, Ho, Wo)
// ---------------------------------------------------------------------------
__global__ void pool2_kernel(const float* __restrict__ in, float* __restrict__ out,
                             int Ho, int Wo) {
    int t = blockIdx.y;
    int p = blockIdx.x * blockDim.x + threadIdx.x;
    if (p >= Ho * Wo) return;
    int h = p / Wo;
    int w = p - h * Wo;
    int Wi = Wo * 2;
    const float* b = in + (size_t)t * (4 * Ho * Wo);
    int i0 = (2 * h) * Wi + 2 * w;
    float v = 0.25f * (b[i0] + b[i0 + 1] + b[i0 + Wi] + b[i0 + Wi + 1]);
    out[(size_t)t * Ho * Wo + p] = v;
}

// ---------------------------------------------------------------------------
// affine_grid + grid_sample (bilinear, zeros padding, align_corners=False)
// img: (T, H, W); theta: (T, 2, 3); writes x_reg: (T, H*W)
// ---------------------------------------------------------------------------
__global__ void sample_kernel(const float* __restrict__ img,
                              const float* __restrict__ theta,
                              float* __restrict__ xreg, int H, int W) {
    int t = blockIdx.y;
    int p = blockIdx.x * blockDim.x + threadIdx.x;
    if (p >= H * W) return;
    int h = p / W;
    int w = p - h * W;

    // pixel-center normalized coords
    float gxn = ((float)w + 0.5f) * (2.0f / (float)W) - 1.0f;
    float gyn = ((float)h + 0.5f) * (2.0f / (float)H) - 1.0f;

    const float* th = theta + t * 6;
    float gx = th[0] * gxn + th[1] * gyn + th[2];
    float gy = th[3] * gxn + th[4] * gyn + th[5];

    // un-normalize (align_corners=False)
    float ix = ((gx + 1.0f) * (float)W - 1.0f) * 0.5f;
    float iy = ((gy + 1.0f) * (float)H - 1.0f) * 0.5f;

    float ix0f = floorf(ix);
    float iy0f = floorf(iy);
    float wx1 = ix - ix0f;
    float wy1 = iy - iy0f;
    int ix0 = (int)ix0f;
    int iy0 = (int)iy0f;

    const float* base = img + (size_t)t * H * W;

    auto g = [&](int yy, int xx) -> float {
        if (yy < 0 || yy >= H || xx < 0 || xx >= W) return 0.0f;
        return base[yy * W + xx];
    };

    float v = g(iy0,     ix0    ) * (1.0f - wy1) * (1.0f - wx1)
            + g(iy0,     ix0 + 1) * (1.0f - wy1) * wx1
            + g(iy0 + 1, ix0    ) * wy1          * (1.0f - wx1)
            + g(iy0 + 1, ix0 + 1) * wy1          * wx1;

    xreg[(size_t)t * H * W + p] = v;
}

// ---------------------------------------------------------------------------
// GEMV via V_WMMA_F32_16X16X4_F32:
//   s[t] += sum_p xreg[t, p] * Wk[p]   for 16 t-rows per block (1 wave).
// A (16x4 f32): lanes 0-15 hold row M=lane, K=0,1; lanes 16-31 hold K=2,3.
// B (4x16 f32): W[p+k] broadcast across all 16 columns -> every D column is
// the same partial dot. Column 0 lives in lanes 0 (rows 0-7) and 16 (rows
// 8-15); flush those with float atomics.
// ---------------------------------------------------------------------------
__global__ void gemv_wmma_kernel(const float* __restrict__ xreg,
                                 const float* __restrict__ Wk,
                                 float* __restrict__ s, int N, int pchunk) {
    int lane   = threadIdx.x;            // blockDim.x == 32, one wave
    int tbase  = blockIdx.y * 16;
    int pstart = blockIdx.x * pchunk;

    int row  = tbase + (lane & 15);
    int koff = (lane >> 4) << 1;         // 0 for lanes 0-15, 2 for lanes 16-31

    const float* arow = xreg + (size_t)row * N;

    v8f c = {0.0f, 0.0f, 0.0f, 0.0f, 0.0f, 0.0f, 0.0f, 0.0f};

#pragma unroll 4
    for (int p = pstart; p < pstart + pchunk; p += 4) {
        v2f a = *(const v2f*)(arow + p + koff);   // 8B-aligned: p%4==0, koff even
        v2f b = *(const v2f*)(Wk   + p + koff);
        c = __builtin_amdgcn_wmma_f32_16x16x4_f32(
                /*neg_a=*/false, a, /*neg_b=*/false, b,
                /*c_mod=*/(short)0, c, /*reuse_a=*/false, /*reuse_b=*/false);
    }

    if (lane == 0) {
#pragma unroll
        for (int j = 0; j < 8; ++j) atomicAdd(&s[tbase + j], c[j]);
    } else if (lane == 16) {
#pragma unroll
        for (int j = 0; j < 8; ++j) atomicAdd(&s[tbase + 8 + j], c[j]);
    }
}

// ---------------------------------------------------------------------------
// rank-1 outer product: L[t, p] = s[t] * Wk[p]
// ---------------------------------------------------------------------------
__global__ void outer_kernel(const float* __restrict__ s,
                             const float* __restrict__ Wk,
                             float* __restrict__ L, int N) {
    int t = blockIdx.y;
    int p = blockIdx.x * blockDim.x + threadIdx.x;
    if (p >= N) return;
    L[(size_t)t * N + p] = s[t] * Wk[p];
}

// ---------------------------------------------------------------------------
extern "C" void kernel_launch(void* const* d_in, const int* in_sizes, int n_in,
                              void* d_out, int out_size, void* d_ws, size_t ws_size,
                              hipStream_t stream) {
    const float* x     = (const float*)d_in[0];  // (256, 262144)
    const float* theta = (const float*)d_in[1];  // (256, 2, 3)
    const float* W1    = (const float*)d_in[2];  // (1, 262144)
    const float* W2    = (const float*)d_in[3];  // (1, 65536)
    const float* W4    = (const float*)d_in[4];  // (1, 16384)
    const float* W8    = (const float*)d_in[5];  // (1, 4096)
    float* out = (float*)d_out;
    float* ws  = (float*)d_ws;

    const int N1 = 512 * 512, N2 = 256 * 256, N4 = 128 * 128, N8 = 64 * 64;

    // workspace: pooled pyramid + dot accumulators (~84 MB + 4 KB)
    float* img2 = ws;
    float* img4 = img2 + (size_t)T_DIM * N2;
    float* img8 = img4 + (size_t)T_DIM * N4;
    float* sacc = img8 + (size_t)T_DIM * N8;     // 4 * 256 floats

    // output layout (tuple flattened in return order)
    float* xr1 = out;
    float* L1  = xr1 + (size_t)T_DIM * N1;
    float* xr2 = L1  + (size_t)T_DIM * N1;
    float* L2  = xr2 + (size_t)T_DIM * N2;
    float* xr4 = L2  + (size_t)T_DIM * N2;
    float* L4  = xr4 + (size_t)T_DIM * N4;
    float* xr8 = L4  + (size_t)T_DIM * N4;
    float* L8  = xr8 + (size_t)T_DIM * N8;

    zero_kernel<<<4, 256, 0, stream>>>(sacc, 4 * T_DIM);

    // pooling pyramid
    pool2_kernel<<<dim3(N2 / 256, T_DIM), 256, 0, stream>>>(x,    img2, 256, 256);
    pool2_kernel<<<dim3(N4 / 256, T_DIM), 256, 0, stream>>>(img2, img4, 128, 128);
    pool2_kernel<<<dim3(N8 / 256, T_DIM), 256, 0, stream>>>(img4, img8,  64,  64);

    // affine grid_sample per level
    sample_kernel<<<dim3(N1 / 256, T_DIM), 256, 0, stream>>>(x,    theta, xr1, 512, 512);
    sample_kernel<<<dim3(N2 / 256, T_DIM), 256, 0, stream>>>(img2, theta, xr2, 256, 256);
    sample_kernel<<<dim3(N4 / 256, T_DIM), 256, 0, stream>>>(img4, theta, xr4, 128, 128);
    sample_kernel<<<dim3(N8 / 256, T_DIM), 256, 0, stream>>>(img8, theta, xr8,  64,  64);

    // WMMA GEMV: s[t] = x_reg[t,:] . W  (128 pixel-chunks x 16 t-tiles, 1 wave each)
    gemv_wmma_kernel<<<dim3(128, 16), 32, 0, stream>>>(xr1, W1, sacc + 0 * 256, N1, N1 / 128);
    gemv_wmma_kernel<<<dim3(128, 16), 32, 0, stream>>>(xr2, W2, sacc + 1 * 256, N2, N2 / 128);
    gemv_wmma_kernel<<<dim3(128, 16), 32, 0, stream>>>(xr4, W4, sacc + 2 * 256, N4, N4 / 128);
    gemv_wmma_kernel<<<dim3(128, 16), 32, 0, stream>>>(xr8, W8, sacc + 3 * 256, N8, N8 / 128);

    // L = s outer W
    outer_kernel<<<dim3(N1 / 256, T_DIM), 256, 0, stream>>>(sacc + 0 * 256, W1, L1, N1);
    outer_kernel<<<dim3(N2 / 256, T_DIM), 256, 0, stream>>>(sacc + 1 * 256, W2, L2, N2);
    outer_kernel<<<dim3(N4 / 256, T_DIM), 256, 0, stream>>>(sacc + 2 * 256, W4, L4, N4);
    outer_kernel<<<dim3(N8 / 256, T_DIM), 256, 0, stream>>>(sacc + 3 * 256, W8, L8, N8);
}